// FLIF_29454885716702
// MI455X (gfx1250) — compile-verified
//
#include <hip/hip_runtime.h>
#include <cmath>

// FLIF layer: S=2048 neurons x T=384 sequential steps.
// History (Toeplitz) convolution done with V_WMMA_F32_16X16X4_F32 on 16-step
// time blocks; only the intra-block tail (<16 MACs/step) is scalar-sequential.
// One wave32 per 16-neuron tile -> 128 workgroups of 32 threads.

typedef __attribute__((ext_vector_type(2))) float v2f;
typedef __attribute__((ext_vector_type(8))) float v8f;

#define S_NEUR 2048
#define T_STEP 384
#define NBLK   (T_STEP / 16)   // 24 time blocks

__global__ __launch_bounds__(32)
void flif_wmma_kernel(const float* __restrict__ I,        // [S, T] row-major
                      const float* __restrict__ weights,  // len = wlen (4999)
                      int wlen,
                      float* __restrict__ spk,            // [S, T]
                      float* __restrict__ trc,            // [S, T]
                      float coef)                         // dt^a * G(2-a) / Cm
{
    // wr[age] = weights[wlen - age], age in 1..T ; wr[0] = 0 (unused slot)
    __shared__ float lds_wr[T_STEP + 1];
    // delta history for this 16-neuron tile: [n][m], m stride 1
    __shared__ float lds_delta[T_STEP * 16];
    // far-history memory for the current block: [m][n_rel]
    __shared__ float lds_far[16 * 16];

    const int lane = threadIdx.x;
    const int m    = lane & 15;   // A-row / B-col / C-col index
    const int hi   = lane >> 4;   // lane half: selects K pair (A/B), M+8 (C)

    for (int a = lane; a <= T_STEP; a += 32)
        lds_wr[a] = (a == 0) ? 0.0f : weights[wlen - a];
    __syncthreads();

    const int row0 = blockIdx.x * 16;
    float V = 0.0f;               // per-neuron V (lanes 0..15 own a neuron)

    for (int b = 0; b < NBLK; ++b) {
        // ---- Phase 1: far history via WMMA (uniform EXEC, all 32 lanes) ----
        // far[m, n] = sum over past blocks bp, K-chunks of 4:
        //             delta[m, k] * wr[(16b + n) - k]
        v8f c = {};
        for (int bp = 0; bp < b; ++bp) {
            #pragma unroll
            for (int ch = 0; ch < 4; ++ch) {
                const int k0 = bp * 16 + ch * 4;
                v2f a, w;
                // A 16x4 f32: lane half -> K {0,1}/{2,3}; VGPR v -> K = v + 2*hi
                a.x = lds_delta[(k0 + 2 * hi    ) * 16 + m];
                a.y = lds_delta[(k0 + 2 * hi + 1) * 16 + m];
                // B 4x16: B[k, n] = wr[(16b + n) - (k0 + k)], k = v + 2*hi
                const int age = b * 16 + m - k0 - 2 * hi;   // age >= 2 here
                w.x = lds_wr[age];
                w.y = lds_wr[age - 1];
                c = __builtin_amdgcn_wmma_f32_16x16x4_f32(
                        false, a, false, w, (short)0, c, false, false);
            }
        }
        // C/D layout: c[v] holds far[M = v + 8*hi][N = m]
        #pragma unroll
        for (int v = 0; v < 8; ++v)
            lds_far[(v + 8 * hi) * 16 + m] = c[v];
        __syncthreads();

        // ---- Phase 2: 16 sequential steps, lanes 0..15 each own a neuron ----
        if (lane < 16) {
            const float* Irow = I   + (size_t)(row0 + lane) * T_STEP;
            float*       srow = spk + (size_t)(row0 + lane) * T_STEP;
            float*       trow = trc + (size_t)(row0 + lane) * T_STEP;
            for (int i = 0; i < 16; ++i) {
                const int n = b * 16 + i;
                // memory = far part + intra-block tail
                float mem = lds_far[lane * 16 + i];
                for (int ip = 0; ip < i; ++ip)
                    mem = fmaf(lds_wr[i - ip],
                               lds_delta[(b * 16 + ip) * 16 + lane], mem);
                const float In = Irow[n];
                // DT/TAU = 0.005, 1/GL = 40, GL = 0.025, VL = -70
                const float V1 = V + 0.005f * (-V + 40.0f * In);
                const float V2 = coef * (-0.025f * (V + 70.0f) + In) + V - mem;
                const float Vpre  = (n == 0) ? -70.0f : ((n == 1) ? V1 : V2);
                const float spike = (V + 50.0f > 0.0f) ? 1.0f : 0.0f; // V>THRESH
                const float Vnew  = Vpre - spike * 20.0f;  // THRESH - V_RESET
                // slot 0 masked out in the reference (j >= 1 only)
                lds_delta[n * 16 + lane] = (n == 0) ? 0.0f : (Vnew - V);
                srow[n] = spike;
                trow[n] = Vnew;
                V = Vnew;
            }
        }
        __syncthreads();   // deltas visible before next block's WMMA phase
    }
}

extern "C" void kernel_launch(void* const* d_in, const int* in_sizes, int n_in,
                              void* d_out, int out_size, void* d_ws, size_t ws_size,
                              hipStream_t stream) {
    const float* I       = (const float*)d_in[0];   // [2048, 384] f32
    const float* weights = (const float*)d_in[1];   // [4999] f32
    const int    wlen    = in_sizes[1];

    float* spk = (float*)d_out;                       // first output, [S, T]
    float* trc = spk + (size_t)S_NEUR * T_STEP;       // second output, [S, T]

    // COEF = dt^alpha * Gamma(2 - alpha) / Cm, computed exactly on host
    const float coef = (float)(pow(0.1, 0.15) * tgamma(2.0 - 0.15) / 0.5);

    flif_wmma_kernel<<<S_NEUR / 16, 32, 0, stream>>>(
        I, weights, wlen, spk, trc, coef);
}